// BEVDeformCrossAttn_10316511445802
// MI455X (gfx1250) — compile-verified
//
#include <hip/hip_runtime.h>
#include <hip/hip_bf16.h>
#include <math.h>

// ---------------- problem constants ----------------
#define KK   6
#define BB   4
#define TT   12
#define DD   256           // DIM == HALF == 256
#define GG   8             // groups / heads
#define BS   (BB*KK*TT)    // 288 query rows (b-major order)
#define NS   (BS*GG)       // 2304 deformable samples
#define HW   200           // BEV H == W
#define PSPAN 102.4f
#define OFFSCALE 4.0f
#define ATTN_SCALE 0.125f  // 64^-0.5

typedef float v2f __attribute__((ext_vector_type(2)));
typedef float v8f __attribute__((ext_vector_type(8)));

// ---------------------------------------------------------------------------
// prep: gather (K,B,T,*) tensors into BS=(b*K+k)*T+t row order; build outmap
// ---------------------------------------------------------------------------
__global__ __launch_bounds__(256)
void prep_kernel(const float* __restrict__ dec_embed,
                 const float* __restrict__ query_scale,
                 const float* __restrict__ ref_points,
                 float* __restrict__ ws_de, float* __restrict__ ws_qs,
                 float* __restrict__ ws_rp, int* __restrict__ ws_map) {
  const int bs = blockIdx.x;           // 0..287
  const int d  = threadIdx.x;          // 0..255
  const int b  = bs / (KK*TT);
  const int r  = bs % (KK*TT);
  const int k  = r / TT;
  const int t  = r % TT;
  const int src = (k*BB + b)*TT + t;   // row in (K,B,T) order
  ws_de[bs*DD + d] = dec_embed  [src*DD + d];
  ws_qs[bs*DD + d] = query_scale[src*DD + d];
  if (d < 2)  ws_rp[bs*2 + d] = ref_points[src*2 + d];
  if (d == 0) ws_map[bs] = src;
}

// ---------------------------------------------------------------------------
// fp32 WMMA GEMM, register-blocked 16x64 per wave (4 C tiles, A reused 4x):
//   C[orow] = act(A @ W + bias) (* mul) (+ addsrc[orow])
// A: MxKd row-major, W: KdxN row-major. V_WMMA_F32_16X16X4_F32 over K.
// Induction pointers -> B loads become immediate-offset loads off one base.
// EXEC stays all-ones (32-thread block, no divergence) as WMMA requires.
// ---------------------------------------------------------------------------
__global__ __launch_bounds__(32)
void gemm_wmma_f32(const float* __restrict__ A, const float* __restrict__ W,
                   const float* __restrict__ bias, const float* __restrict__ mul,
                   const float* __restrict__ addsrc, const int* __restrict__ outmap,
                   float* __restrict__ C, int M, int N, int Kd, int act) {
  const int stripsN = N >> 6;          // 64-wide output strips
  const int tm = blockIdx.x / stripsN;
  const int tn = blockIdx.x % stripsN; // strip index (64 cols)
  const int lane = threadIdx.x;        // 0..31
  const int lo = lane & 15;
  const int hi = lane >> 4;            // half-wave select (K lo/hi pair)

  // A-matrix fragment source: row M=lo, K pair (hi*2, hi*2+1)
  const float* ap = A + (size_t)(tm*16 + lo)*Kd + hi*2;
  // B-matrix fragment base: col tn*64+lo (+ j*16 per tile), rows k0/k0+1
  const float* wp = W + (size_t)(hi*2)*N + tn*64 + lo;

  v8f c0 = {}, c1 = {}, c2 = {}, c3 = {};
  #pragma unroll 4
  for (int kk = 0; kk < Kd; kk += 4) {
    const v2f a = *(const v2f*)ap;     // one A fragment, reused by 4 WMMAs
    v2f b0, b1, b2, b3;
    b0.x = wp[0];      b0.y = wp[N + 0];
    b1.x = wp[16];     b1.y = wp[N + 16];
    b2.x = wp[32];     b2.y = wp[N + 32];
    b3.x = wp[48];     b3.y = wp[N + 48];
    c0 = __builtin_amdgcn_wmma_f32_16x16x4_f32(false, a, false, b0, (short)0, c0, false, false);
    c1 = __builtin_amdgcn_wmma_f32_16x16x4_f32(false, a, false, b1, (short)0, c1, false, false);
    c2 = __builtin_amdgcn_wmma_f32_16x16x4_f32(false, a, false, b2, (short)0, c2, false, false);
    c3 = __builtin_amdgcn_wmma_f32_16x16x4_f32(false, a, false, b3, (short)0, c3, false, false);
    ap += 4;                           // next K quad
    wp += (size_t)4*N;
  }

  // C/D layout: VGPR r -> M = r (lanes 0-15) or r+8 (lanes 16-31), N = lane%16
  v8f accs[4] = {c0, c1, c2, c3};
  #pragma unroll
  for (int j = 0; j < 4; ++j) {
    #pragma unroll
    for (int r = 0; r < 8; ++r) {
      const int m = tm*16 + r + hi*8;
      const int n = tn*64 + j*16 + lo;
      float v = accs[j][r];
      if (bias)     v += bias[n];
      if (act == 1) v = fmaxf(v, 0.0f);
      if (mul)      v *= mul[(size_t)m*N + n];
      const int orow = outmap ? outmap[m] : m;
      if (addsrc)   v += addsrc[(size_t)orow*N + n];
      C[(size_t)orow*N + n] = v;
    }
  }
}

// ---------------------------------------------------------------------------
// offset MLP: per (query, group): gelu(cq@Wo1+bo1)@Wo2+bo2 -> tanh*4 + rp
// ---------------------------------------------------------------------------
__global__ __launch_bounds__(256)
void offsets_kernel(const float* __restrict__ conq, const float* __restrict__ rp,
                    const float* __restrict__ Wo1, const float* __restrict__ bo1,
                    const float* __restrict__ Wo2, const float* __restrict__ bo2,
                    float* __restrict__ vgrid) {
  const int t = blockIdx.x*blockDim.x + threadIdx.x;
  if (t >= NS) return;
  const int bs = t >> 3;
  const int g  = t & 7;
  const float* cq = conq + (size_t)bs*DD + g*32;
  float x[32];
  #pragma unroll
  for (int i = 0; i < 32; ++i) x[i] = cq[i];
  float o0 = bo2[0], o1 = bo2[1];
  for (int j = 0; j < 64; ++j) {
    float h = bo1[j];
    #pragma unroll
    for (int i = 0; i < 32; ++i) h = fmaf(x[i], Wo1[i*64 + j], h);
    h = 0.5f * h * (1.0f + erff(h * 0.70710678118654752f));   // exact gelu
    o0 = fmaf(h, Wo2[j*2 + 0], o0);
    o1 = fmaf(h, Wo2[j*2 + 1], o1);
  }
  vgrid[t*2 + 0] = rp[bs*2 + 0] + tanhf(o0) * OFFSCALE;       // meters
  vgrid[t*2 + 1] = rp[bs*2 + 1] + tanhf(o1) * OFFSCALE;
}

// ---------------------------------------------------------------------------
// bilinear gather of RAW BEV features at the 2304 sample points.
// (conv1x1 has no bias => it commutes with grid_sample; GEMM afterwards.)
// one block per sample, one thread per channel.
// ---------------------------------------------------------------------------
__global__ __launch_bounds__(256)
void sample_kernel(const float* __restrict__ bev, const float* __restrict__ vgrid,
                   float* __restrict__ sfeat) {
  const int s = blockIdx.x;            // 0..2303  (b, kt, g) order
  const int c = threadIdx.x;           // channel
  const int b = s / (KK*TT*GG);
  const float vx = vgrid[s*2 + 0];
  const float vy = vgrid[s*2 + 1];
  const float gx =  2.0f * vx / PSPAN;                 // normalize_grid
  const float gy = -2.0f * vy / PSPAN;                 // flipped y
  const float px = (gx + 1.0f) * (HW * 0.5f) - 0.5f;   // align_corners=False
  const float py = (gy + 1.0f) * (HW * 0.5f) - 0.5f;
  const float x0f = floorf(px), y0f = floorf(py);
  const int   x0 = (int)x0f,    y0 = (int)y0f;
  const float wx1 = px - x0f, wx0 = 1.0f - wx1;
  const float wy1 = py - y0f, wy0 = 1.0f - wy1;
  const float* base = bev + ((size_t)b*DD + c) * (HW*HW);
  float acc = 0.0f;
  #pragma unroll
  for (int dy = 0; dy < 2; ++dy) {
    #pragma unroll
    for (int dx = 0; dx < 2; ++dx) {
      const int ix = x0 + dx, iy = y0 + dy;
      const float w = (dx ? wx1 : wx0) * (dy ? wy1 : wy0);
      if (ix >= 0 && ix < HW && iy >= 0 && iy < HW)    // zeros padding
        acc = fmaf(w, base[(size_t)iy*HW + ix], acc);
    }
  }
  sfeat[(size_t)s*DD + c] = acc;
}

// ---------------------------------------------------------------------------
// DAB-DETR sine embedding: out = cat(emb(pos.y), emb(pos.x)), 256 dims
// ---------------------------------------------------------------------------
__global__ __launch_bounds__(256)
void sine_kernel(const float* __restrict__ pos, float* __restrict__ out, int rows) {
  const int r = blockIdx.x;
  const int j = threadIdx.x;           // 0..255
  const float p  = (j < 128) ? pos[r*2 + 1] : pos[r*2 + 0];
  const int   jj = j & 127;
  const int   i  = jj >> 1;
  // 1 / dim_t = 10000^(-i/64)
  const float inv = expf(-(float)i * (9.210340371976184f / 64.0f));
  const float th  = p * 6.283185307179586f * inv;
  out[(size_t)r*DD + j] = (jj & 1) ? cosf(th) : sinf(th);
}

// ---------------------------------------------------------------------------
// attention: one wave per (query, head); lane d covers con-dim d & pos-dim d.
// sim_g via full-wave __shfl_xor reduction, softmax over G=8, weighted V sum.
// ---------------------------------------------------------------------------
__global__ __launch_bounds__(32)
void attn_kernel(const float* __restrict__ conq, const float* __restrict__ posq,
                 const float* __restrict__ conk, const float* __restrict__ posk,
                 const float* __restrict__ vv,   float* __restrict__ outp) {
  const int bs = blockIdx.x >> 3;
  const int h  = blockIdx.x & 7;
  const int d  = threadIdx.x;          // 0..31
  const int qo = bs*DD + h*32 + d;
  const float q0 = conq[qo] * ATTN_SCALE;
  const float q1 = posq[qo] * ATTN_SCALE;
  float sim[GG];
  #pragma unroll
  for (int g = 0; g < GG; ++g) {
    const int ko = (bs*GG + g)*DD + h*32 + d;
    float p = q0 * conk[ko] + q1 * posk[ko];
    #pragma unroll
    for (int m = 16; m >= 1; m >>= 1) p += __shfl_xor(p, m, 32);
    sim[g] = p;                        // full dot, all lanes hold it
  }
  float mx = sim[0];
  #pragma unroll
  for (int g = 1; g < GG; ++g) mx = fmaxf(mx, sim[g]);
  float se = 0.0f;
  #pragma unroll
  for (int g = 0; g < GG; ++g) { sim[g] = expf(sim[g] - mx); se += sim[g]; }
  const float inv = 1.0f / se;
  float o = 0.0f;
  #pragma unroll
  for (int g = 0; g < GG; ++g)
    o = fmaf(sim[g] * inv, vv[(size_t)(bs*GG + g)*DD + h*32 + d], o);
  outp[qo] = o;
}

// ---------------------------------------------------------------------------
extern "C" void kernel_launch(void* const* d_in, const int* in_sizes, int n_in,
                              void* d_out, int out_size, void* d_ws, size_t ws_size,
                              hipStream_t stream) {
  const float* dec_embed = (const float*)d_in[0];
  const float* bev_feat  = (const float*)d_in[1];
  const float* qscale    = (const float*)d_in[2];
  const float* refpts    = (const float*)d_in[3];
  const float* W_con_q   = (const float*)d_in[4];
  const float* b_con_q   = (const float*)d_in[5];
  const float* W_con_k   = (const float*)d_in[6];
  const float* W_v       = (const float*)d_in[7];
  const float* Wq1 = (const float*)d_in[8];  const float* bq1 = (const float*)d_in[9];
  const float* Wq2 = (const float*)d_in[10]; const float* bq2 = (const float*)d_in[11];
  const float* Wk1 = (const float*)d_in[12]; const float* bk1 = (const float*)d_in[13];
  const float* Wk2 = (const float*)d_in[14]; const float* bk2 = (const float*)d_in[15];
  const float* Wo1 = (const float*)d_in[16]; const float* bo1 = (const float*)d_in[17];
  const float* Wo2 = (const float*)d_in[18]; const float* bo2 = (const float*)d_in[19];
  const float* W_out = (const float*)d_in[20]; const float* b_out = (const float*)d_in[21];
  float* out = (float*)d_out;

  // -------- workspace carve-up (~16.3 MB of fp32) --------
  float* w = (float*)d_ws;
  float* ws_de   = w; w += BS*DD;
  float* ws_qs   = w; w += BS*DD;
  float* ws_rp   = w; w += BS*2;
  float* ws_conq = w; w += BS*DD;
  float* ws_qse  = w; w += BS*DD;
  float* ws_tmpq = w; w += BS*DD;
  float* ws_posq = w; w += BS*DD;
  float* ws_vg   = w; w += NS*2;
  float* ws_s    = w; w += NS*DD;
  float* ws_kse  = w; w += NS*DD;
  float* ws_tmpk = w; w += NS*DD;
  float* ws_posk = w; w += NS*DD;
  float* ws_conk = w; w += NS*DD;
  float* ws_v    = w; w += NS*DD;
  float* ws_att  = w; w += BS*DD;
  int*   ws_map  = (int*)w;

  const int stripsQ = (BS/16)*(DD/64);   // 72 wave-strips for 288-row GEMMs
  const int stripsS = (NS/16)*(DD/64);   // 576 wave-strips for 2304-row GEMMs

  // 1) row reorder of (K,B,T,*) inputs + output row map
  prep_kernel<<<BS, 256, 0, stream>>>(dec_embed, qscale, refpts,
                                      ws_de, ws_qs, ws_rp, ws_map);
  // 2) con_q = de @ W_con_q + b
  gemm_wmma_f32<<<stripsQ, 32, 0, stream>>>(ws_de, W_con_q, b_con_q,
      nullptr, nullptr, nullptr, ws_conq, BS, DD, DD, 0);
  // 3) offset MLP -> sample points (meters)
  offsets_kernel<<<NS/256, 256, 0, stream>>>(ws_conq, ws_rp, Wo1, bo1, Wo2, bo2, ws_vg);
  // 4) bilinear gather of raw BEV features (conv commutes with sampling)
  sample_kernel<<<NS, 256, 0, stream>>>(bev_feat, ws_vg, ws_s);
  // 5) sine embeddings
  sine_kernel<<<BS, 256, 0, stream>>>(ws_rp, ws_qse, BS);
  sine_kernel<<<NS, 256, 0, stream>>>(ws_vg, ws_kse, NS);
  // 6) positional MLPs + sampled-feature projections (all WMMA GEMMs)
  gemm_wmma_f32<<<stripsQ, 32, 0, stream>>>(ws_qse, Wq1, bq1,
      nullptr, nullptr, nullptr, ws_tmpq, BS, DD, DD, 1);
  gemm_wmma_f32<<<stripsQ, 32, 0, stream>>>(ws_tmpq, Wq2, bq2,
      ws_qs, nullptr, nullptr, ws_posq, BS, DD, DD, 0);
  gemm_wmma_f32<<<stripsS, 32, 0, stream>>>(ws_kse, Wk1, bk1,
      nullptr, nullptr, nullptr, ws_tmpk, NS, DD, DD, 1);
  gemm_wmma_f32<<<stripsS, 32, 0, stream>>>(ws_tmpk, Wk2, bk2,
      nullptr, nullptr, nullptr, ws_posk, NS, DD, DD, 0);
  gemm_wmma_f32<<<stripsS, 32, 0, stream>>>(ws_s, W_con_k, nullptr,
      nullptr, nullptr, nullptr, ws_conk, NS, DD, DD, 0);
  gemm_wmma_f32<<<stripsS, 32, 0, stream>>>(ws_s, W_v, nullptr,
      nullptr, nullptr, nullptr, ws_v, NS, DD, DD, 0);
  // 7) 8-way deformable attention per (query, head)
  attn_kernel<<<BS*GG, 32, 0, stream>>>(ws_conq, ws_posq, ws_conk, ws_posk,
                                        ws_v, ws_att);
  // 8) out = identity + attn @ W_out + b_out, remapped to (K,B,T) rows
  gemm_wmma_f32<<<stripsQ, 32, 0, stream>>>(ws_att, W_out, b_out,
      nullptr, dec_embed, ws_map, out, BS, DD, DD, 0);
}